// Encoder_23605140259289
// MI455X (gfx1250) — compile-verified
//
#include <hip/hip_runtime.h>
#include <hip/hip_bf16.h>

// ---------------- problem constants (from reference) ----------------
#define N_NODES 50000
#define N_EDGES 1600000
#define IN_CH   9
#define EMB_DIM 192
#define K1      200      // IN_FEAT = 9 + 192 - 1
#define HID     128
#define OUTF    64

typedef float v2f __attribute__((ext_vector_type(2)));
typedef float v8f __attribute__((ext_vector_type(8)));

// ---------------- degree / dinv prep ----------------
__global__ void zero_f32_kernel(float* __restrict__ p, int n) {
    int i = blockIdx.x * blockDim.x + threadIdx.x;
    if (i < n) p[i] = 0.0f;
}

__global__ void deg_count_kernel(const int* __restrict__ dst, float* __restrict__ deg, int e) {
    int i = blockIdx.x * blockDim.x + threadIdx.x;
    if (i < e) atomicAdd(&deg[dst[i]], 1.0f);
}

__global__ void dinv_finalize_kernel(float* __restrict__ deg, int n) {
    int i = blockIdx.x * blockDim.x + threadIdx.x;
    if (i < n) deg[i] = rsqrtf(deg[i] + 1.0f);
}

// ---------------- weight repack: pair-major B layout for WMMA ----------------
// Wp[(kp*NOUT + col)*2 + {0,1}] = { W[2kp][col], W[2kp+1][col] }
// -> a single global_load_b64 yields the (K, K+1) pair each lane needs.
template <int KIN, int NOUT>
__global__ void repack_pairs_kernel(const float* __restrict__ W, float* __restrict__ Wp) {
    int i = blockIdx.x * blockDim.x + threadIdx.x;
    if (i >= (KIN / 2) * NOUT) return;
    int kp  = i / NOUT;
    int col = i - kp * NOUT;
    Wp[i * 2 + 0] = W[(2 * kp)     * NOUT + col];
    Wp[i * 2 + 1] = W[(2 * kp + 1) * NOUT + col];
}

// ---------------- layer-1 GEMM with fused one-hot|embedding gather ----------------
// hw  = [one_hot | emb[idx]] @ W1  (bias applied post-aggregation)
// agg = hw * dinv^2 + b            (self-loop term + bias, ready for edge scatter)
// One block = 16 rows; wave w -> 16-col tile w. 8 waves * 16 = 128 cols. No bounds
// checks: N divisible by 16, so EXEC stays all-ones through every WMMA (ISA req).
__global__ __launch_bounds__(256)
void gemm1_wmma_kernel(const float* __restrict__ x, const float* __restrict__ emb,
                       const float* __restrict__ Wp, const float* __restrict__ bias,
                       const float* __restrict__ dinv,
                       float* __restrict__ hw, float* __restrict__ agg) {
    const int lane    = threadIdx.x & 31;
    const int wave    = threadIdx.x >> 5;
    const int half    = lane >> 4;        // 0: K,K+1 lanes | 1: K+2,K+3 lanes
    const int l16     = lane & 15;
    const int rowBase = blockIdx.x * 16;
    const int col     = wave * 16 + l16;
    const int arow    = rowBase + l16;    // A row (M = l16 in both halves)

    const float* xRow = x + arow * IN_CH;
    const int    idx  = (int)xRow[IN_CH - 1];
    const float* bp   = Wp + 2 * (half * HID + col);   // packed B base (8B aligned)

    v8f c = {};
    // k0 = 0, 4: features come from x (only 4B-aligned rows -> scalar loads)
    for (int k0 = 0; k0 < 8; k0 += 4) {
        const int ka = k0 + 2 * half;
        v2f a, b;
        a.x = xRow[ka];
        a.y = xRow[ka + 1];
        b = *(const v2f*)(bp + k0 * HID);
        c = __builtin_amdgcn_wmma_f32_16x16x4_f32(false, a, false, b, (short)0, c, false, false);
    }
    // k0 = 8..196: features come from the embedding row (8B-aligned v2f loads)
    const float* eBase = emb + idx * EMB_DIM - 8 + 2 * half;
    for (int k0 = 8; k0 < K1; k0 += 4) {
        v2f a = *(const v2f*)(eBase + k0);
        v2f b = *(const v2f*)(bp + k0 * HID);
        c = __builtin_amdgcn_wmma_f32_16x16x4_f32(false, a, false, b, (short)0, c, false, false);
    }

    const float bc = bias[col];
#pragma unroll
    for (int v = 0; v < 8; ++v) {        // C/D layout: lane<16 -> M=v, lane>=16 -> M=v+8
        const int r    = rowBase + v + 8 * half;
        const float val = c[v];
        hw[r * HID + col] = val;
        const float di = dinv[r];
        agg[r * HID + col] = val * di * di + bc;
    }
}

// ---------------- generic dense GEMM (layers 2..4), packed-B ----------------
template <int KIN, int NOUT>
__global__ __launch_bounds__(256)
void gemm_wmma_kernel(const float* __restrict__ Xin, const float* __restrict__ Wp,
                      const float* __restrict__ bias, const float* __restrict__ dinv,
                      float* __restrict__ hw, float* __restrict__ agg) {
    const int lane    = threadIdx.x & 31;
    const int wave    = threadIdx.x >> 5;
    const int half    = lane >> 4;
    const int l16     = lane & 15;
    const int rowBase = blockIdx.x * 16;
    const int col     = wave * 16 + l16;
    const int arow    = rowBase + l16;

    const float* aBase = Xin + arow * KIN + 2 * half;  // 8B aligned (KIN mult of 2, row*KIN even)
    const float* bp    = Wp + 2 * (half * NOUT + col);

    v8f c = {};
    for (int k0 = 0; k0 < KIN; k0 += 4) {
        v2f a = *(const v2f*)(aBase + k0);
        v2f b = *(const v2f*)(bp + k0 * NOUT);
        c = __builtin_amdgcn_wmma_f32_16x16x4_f32(false, a, false, b, (short)0, c, false, false);
    }

    const float bc = bias[col];
#pragma unroll
    for (int v = 0; v < 8; ++v) {
        const int r    = rowBase + v + 8 * half;
        const float val = c[v];
        hw[r * NOUT + col] = val;
        const float di = dinv[r];
        agg[r * NOUT + col] = val * di * di + bc;
    }
}

// ---------------- edge scatter: agg[dst] += hw[src] * dinv[src]*dinv[dst] ----------------
// One wave32 per edge; lane covers F/32 consecutive floats -> one b128/b64 gather per
// lane, then scalar f32 atomics. hw/agg are 25.6MB: L2-resident (192MB L2), so this
// random traffic never hits HBM.
template <int F>
__global__ __launch_bounds__(256)
void scatter_add_kernel(const float* __restrict__ hw, float* __restrict__ agg,
                        const int* __restrict__ src, const int* __restrict__ dst,
                        const float* __restrict__ dinv, int e) {
    const int lane = threadIdx.x & 31;
    const int wave = threadIdx.x >> 5;
    const int edge = blockIdx.x * 8 + wave;
    if (edge >= e) return;
    const int s = src[edge];
    const int d = dst[edge];
    const float coef = dinv[s] * dinv[d];
    constexpr int FPL = F / 32;
    const float* hs = hw  + s * F + lane * FPL;
    float*       ad = agg + d * F + lane * FPL;
    if constexpr (FPL == 4) {
        const float4 hv = *(const float4*)hs;   // global_load_b128
        atomicAdd(&ad[0], hv.x * coef);
        atomicAdd(&ad[1], hv.y * coef);
        atomicAdd(&ad[2], hv.z * coef);
        atomicAdd(&ad[3], hv.w * coef);
    } else {
        const float2 hv = *(const float2*)hs;   // global_load_b64
        atomicAdd(&ad[0], hv.x * coef);
        atomicAdd(&ad[1], hv.y * coef);
    }
}

__global__ void relu_kernel(float* __restrict__ p, int n) {
    int i = blockIdx.x * blockDim.x + threadIdx.x;
    if (i < n) p[i] = fmaxf(p[i], 0.0f);
}

// ---------------- host-side launcher ----------------
static inline size_t align256(size_t v) { return (v + 255) & ~size_t(255); }

extern "C" void kernel_launch(void* const* d_in, const int* in_sizes, int n_in,
                              void* d_out, int out_size, void* d_ws, size_t ws_size,
                              hipStream_t stream) {
    const float* x   = (const float*)d_in[0];
    const int*   ei  = (const int*)  d_in[1];   // (2,E): src = ei, dst = ei+E
    const float* emb = (const float*)d_in[2];
    const float* W1  = (const float*)d_in[3];
    const float* b1  = (const float*)d_in[4];
    const float* W2  = (const float*)d_in[5];
    const float* b2  = (const float*)d_in[6];
    const float* W3  = (const float*)d_in[7];
    const float* b3  = (const float*)d_in[8];
    const float* W4  = (const float*)d_in[9];
    const float* b4  = (const float*)d_in[10];
    float* out = (float*)d_out;                 // N x 64

    const int* srcI = ei;
    const int* dstI = ei + N_EDGES;

    // workspace carve-up
    char* ws = (char*)d_ws;
    float* dinv = (float*)ws;  ws += align256(sizeof(float) * N_NODES);
    float* hw   = (float*)ws;  ws += align256(sizeof(float) * N_NODES * HID);
    float* aggA = (float*)ws;  ws += align256(sizeof(float) * N_NODES * HID);
    float* aggB = (float*)ws;  ws += align256(sizeof(float) * N_NODES * HID);
    float* Wp1  = (float*)ws;  ws += align256(sizeof(float) * K1  * HID);
    float* Wp2  = (float*)ws;  ws += align256(sizeof(float) * HID * HID);
    float* Wp3  = (float*)ws;  ws += align256(sizeof(float) * HID * HID);
    float* Wp4  = (float*)ws;  ws += align256(sizeof(float) * HID * OUTF);

    const int TB = 256;
    const dim3 gN((N_NODES + TB - 1) / TB);
    const dim3 gE((N_EDGES + TB - 1) / TB);
    const dim3 gTiles(N_NODES / 16);            // 3125 exact
    const dim3 gEdgeWaves((N_EDGES + 7) / 8);   // 8 edges (waves) per block
    const dim3 gRelu((N_NODES * HID + TB - 1) / TB);

    // weight repack (tiny, once per call)
    repack_pairs_kernel<K1,  HID ><<<dim3((K1/2  * HID  + TB-1)/TB), TB, 0, stream>>>(W1, Wp1);
    repack_pairs_kernel<HID, HID ><<<dim3((HID/2 * HID  + TB-1)/TB), TB, 0, stream>>>(W2, Wp2);
    repack_pairs_kernel<HID, HID ><<<dim3((HID/2 * HID  + TB-1)/TB), TB, 0, stream>>>(W3, Wp3);
    repack_pairs_kernel<HID, OUTF><<<dim3((HID/2 * OUTF + TB-1)/TB), TB, 0, stream>>>(W4, Wp4);

    // degree -> dinv = rsqrt(deg + 1)
    zero_f32_kernel<<<gN, TB, 0, stream>>>(dinv, N_NODES);
    deg_count_kernel<<<gE, TB, 0, stream>>>(dstI, dinv, N_EDGES);
    dinv_finalize_kernel<<<gN, TB, 0, stream>>>(dinv, N_NODES);

    // ---- layer 1: fused embed-gather GEMM (K=200) ----
    gemm1_wmma_kernel<<<gTiles, 256, 0, stream>>>(x, emb, Wp1, b1, dinv, hw, aggA);
    scatter_add_kernel<HID><<<gEdgeWaves, 256, 0, stream>>>(hw, aggA, srcI, dstI, dinv, N_EDGES);
    relu_kernel<<<gRelu, TB, 0, stream>>>(aggA, N_NODES * HID);

    // ---- layer 2 ----
    gemm_wmma_kernel<HID, HID><<<gTiles, 256, 0, stream>>>(aggA, Wp2, b2, dinv, hw, aggB);
    scatter_add_kernel<HID><<<gEdgeWaves, 256, 0, stream>>>(hw, aggB, srcI, dstI, dinv, N_EDGES);
    relu_kernel<<<gRelu, TB, 0, stream>>>(aggB, N_NODES * HID);

    // ---- layer 3 ----
    gemm_wmma_kernel<HID, HID><<<gTiles, 256, 0, stream>>>(aggB, Wp3, b3, dinv, hw, aggA);
    scatter_add_kernel<HID><<<gEdgeWaves, 256, 0, stream>>>(hw, aggA, srcI, dstI, dinv, N_EDGES);
    relu_kernel<<<gRelu, TB, 0, stream>>>(aggA, N_NODES * HID);

    // ---- layer 4: output 64-wide, no relu, agg written straight into d_out ----
    gemm_wmma_kernel<HID, OUTF><<<gTiles, 128, 0, stream>>>(aggA, Wp4, b4, dinv, hw, out);
    scatter_add_kernel<OUTF><<<gEdgeWaves, 256, 0, stream>>>(hw, out, srcI, dstI, dinv, N_EDGES);
}